// SCMPALoss_76149770158663
// MI455X (gfx1250) — compile-verified
//
#include <hip/hip_runtime.h>
#include <hip/hip_bf16.h>

// SCMPA loss on MI455X (gfx1250, wave32).
// Stage 1 (HBM-bound): per-row dot & squared norms via diag accumulation with
//   V_WMMA_F32_16X16X4_F32 (full f32 precision). D split into KSPLIT slices for
//   occupancy (256 blocks / 2048 waves); tiles staged memory->LDS with the CDNA5
//   async path (GLOBAL_LOAD_ASYNC_TO_LDS_B128 + s_wait_asynccnt).
// Stage 1b: combine slice partials -> sim[4096].
// Stage 2: 4096x4096 pairwise hinge, 2-D block decomposition, deterministic partials.
// Stage 3: finalize (+ Huber) -> out[0..2] = (total, L_c, L_s).

#define USE_ASYNC_LDS 1

typedef float v2f __attribute__((ext_vector_type(2)));
typedef float v8f __attribute__((ext_vector_type(8)));

#define NROWS 4096
#define DIMS 2048
#define MARGIN_F 0.05f
#define HUBER_DELTA_F 0.1f
#define COS_EPS_F 1e-8f

#define KSPLIT 8                        // D-dimension slices (occupancy)
#define COLS_PER_WAVE (DIMS / KSPLIT)   // 256
#define TILE_K 32
#define LDS_STRIDE 36                   // 32 cols + 4 pad -> conflict-free b64 lane reads
#define WAVES_PER_BLOCK 8
#define ROWS_PER_WAVE 16

// ---------------------------------------------------------------------------
// Stage 1: per-(slice,row) partials dot/na2/nb2 via WMMA f32 diag accumulation.
// One wave32 owns 16 rows x 256 cols. Per-wave LDS tiles -> no barriers needed.
// ---------------------------------------------------------------------------
__global__ __launch_bounds__(256) void scmpa_partials_wmma(const float* __restrict__ zref,
                                                           const float* __restrict__ zperf,
                                                           float* __restrict__ pdot,
                                                           float* __restrict__ pna,
                                                           float* __restrict__ pnb) {
    __shared__ float lds[WAVES_PER_BLOCK][2][ROWS_PER_WAVE * LDS_STRIDE];

    const int lane   = threadIdx.x & 31;
    const int wave   = threadIdx.x >> 5;
    const int slice  = blockIdx.x & (KSPLIT - 1);
    const int rowBlk = blockIdx.x >> 3;                 // log2(KSPLIT)
    const int rowBase = (rowBlk * WAVES_PER_BLOCK + wave) * ROWS_PER_WAVE;
    const int kStart  = slice * COLS_PER_WAVE;

    float* tileA = &lds[wave][0][0];
    float* tileB = &lds[wave][1][0];
#if USE_ASYNC_LDS
    // Low 32 bits of a flat shared pointer == LDS byte offset (LDS aperture).
    const unsigned ldsA = (unsigned)(size_t)tileA;
    const unsigned ldsB = (unsigned)(size_t)tileB;
#endif

    v8f accD = {};   // diag -> per-row dot(zref, zperf)
    v8f accA = {};   // diag -> per-row |zref|^2
    v8f accB = {};   // diag -> per-row |zperf|^2

    const int half = lane >> 4;   // K pair select {0,1} vs {2,3}
    const int lrow = lane & 15;   // A-row / B-col owned by this lane

    for (int k0 = 0; k0 < COLS_PER_WAVE; k0 += TILE_K) {
#if USE_ASYNC_LDS
        // previous tile's ds_loads must finish reading before async writes land
        asm volatile("s_wait_dscnt 0x0" ::: "memory");
#pragma unroll
        for (int i = 0; i < 4; ++i) {
            const int t  = i * 32 + lane;               // 16 rows x 8 float4 cols
            const int r  = t >> 3;
            const int c4 = (t & 7) * 4;
            const unsigned      loff = (unsigned)((r * LDS_STRIDE + c4) * 4);
            const size_t        gidx = (size_t)(rowBase + r) * DIMS +
                                       (size_t)(kStart + k0 + c4);
            const unsigned long long ga = (unsigned long long)(zref + gidx);
            const unsigned long long gb = (unsigned long long)(zperf + gidx);
            const unsigned la = ldsA + loff;
            const unsigned lb = ldsB + loff;
            asm volatile("global_load_async_to_lds_b128 %0, %1, off"
                         :: "v"(la), "v"(ga) : "memory");
            asm volatile("global_load_async_to_lds_b128 %0, %1, off"
                         :: "v"(lb), "v"(gb) : "memory");
        }
        asm volatile("s_wait_asynccnt 0x0" ::: "memory");
#else
#pragma unroll
        for (int i = 0; i < 4; ++i) {
            const int t  = i * 32 + lane;
            const int r  = t >> 3;
            const int c4 = (t & 7) * 4;
            const size_t gidx = (size_t)(rowBase + r) * DIMS + (size_t)(kStart + k0 + c4);
            const float4 va = *(const float4*)(zref + gidx);
            const float4 vb = *(const float4*)(zperf + gidx);
            *(float4*)(tileA + r * LDS_STRIDE + c4) = va;
            *(float4*)(tileB + r * LDS_STRIDE + c4) = vb;
        }
        __syncthreads();
#endif

        // 8 WMMA K-steps of 4; the same lane registers serve as A and as B.
#pragma unroll
        for (int k = 0; k < TILE_K / 4; ++k) {
            const int c0 = k * 4 + half * 2;
            v2f a, b;
            a.x = tileA[lrow * LDS_STRIDE + c0];
            a.y = tileA[lrow * LDS_STRIDE + c0 + 1];
            b.x = tileB[lrow * LDS_STRIDE + c0];
            b.y = tileB[lrow * LDS_STRIDE + c0 + 1];
            accD = __builtin_amdgcn_wmma_f32_16x16x4_f32(false, a, false, b,
                                                         (short)0, accD, false, false);
            accA = __builtin_amdgcn_wmma_f32_16x16x4_f32(false, a, false, a,
                                                         (short)0, accA, false, false);
            accB = __builtin_amdgcn_wmma_f32_16x16x4_f32(false, b, false, b,
                                                         (short)0, accB, false, false);
        }
#if !USE_ASYNC_LDS
        __syncthreads();
#endif
    }

    // diag m (0..7): lane m, elem m;  diag m (8..15): lane m+16, elem m-8
    int diagIdx = -1;
    int row = 0;
    if (lane < 8) {
        diagIdx = lane;
        row = rowBase + lane;
    } else if (lane >= 24) {
        diagIdx = lane - 24;
        row = rowBase + 8 + (lane - 24);
    }
    if (diagIdx >= 0) {
        float d = 0.0f, na2 = 0.0f, nb2 = 0.0f;
#pragma unroll
        for (int j = 0; j < 8; ++j) {
            d   = (j == diagIdx) ? accD[j] : d;
            na2 = (j == diagIdx) ? accA[j] : na2;
            nb2 = (j == diagIdx) ? accB[j] : nb2;
        }
        const size_t o = (size_t)slice * NROWS + (size_t)row;
        pdot[o] = d;
        pna[o]  = na2;
        pnb[o]  = nb2;
    }
}

// ---------------------------------------------------------------------------
// Stage 1b: combine KSPLIT slice partials -> sim[row].
// ---------------------------------------------------------------------------
__global__ __launch_bounds__(256) void scmpa_sim_combine(const float* __restrict__ pdot,
                                                         const float* __restrict__ pna,
                                                         const float* __restrict__ pnb,
                                                         float* __restrict__ sim) {
    const int r = blockIdx.x * 256 + threadIdx.x;
    float d = 0.0f, na2 = 0.0f, nb2 = 0.0f;
#pragma unroll
    for (int s = 0; s < KSPLIT; ++s) {
        d   += pdot[(size_t)s * NROWS + r];
        na2 += pna[(size_t)s * NROWS + r];
        nb2 += pnb[(size_t)s * NROWS + r];
    }
    const float na = fmaxf(sqrtf(na2), COS_EPS_F);
    const float nb = fmaxf(sqrtf(nb2), COS_EPS_F);
    sim[r] = d / (na * nb);
}

// ---------------------------------------------------------------------------
// Stage 2: masked pairwise hinge sums. 256 blocks (16 i-blocks x 16 j-blocks),
// j-tile in LDS; per-block deterministic partials (no float atomics).
// ---------------------------------------------------------------------------
__global__ __launch_bounds__(256) void scmpa_pairwise(const float* __restrict__ sim,
                                                      const float* __restrict__ tgt,
                                                      float* __restrict__ partial_sum,
                                                      float* __restrict__ partial_cnt) {
    __shared__ float s_sim[256];
    __shared__ float s_ov[256];
    __shared__ float red[256];

    const int bi = blockIdx.x >> 4;
    const int bj = blockIdx.x & 15;
    const int i  = bi * 256 + threadIdx.x;
    const int jj = bj * 256 + threadIdx.x;

    const float sim_i = sim[i];
    const float ov_i  = tgt[(size_t)i * 5 + 4];
    s_sim[threadIdx.x] = sim[jj];
    s_ov[threadIdx.x]  = tgt[(size_t)jj * 5 + 4];
    __syncthreads();

    float acc = 0.0f;
    float cnt = 0.0f;
#pragma unroll 8
    for (int j = 0; j < 256; ++j) {
        const bool  m  = ov_i > s_ov[j];
        const float pl = fmaxf(s_sim[j] - sim_i + MARGIN_F, 0.0f);
        acc += m ? pl : 0.0f;
        cnt += m ? 1.0f : 0.0f;
    }

    red[threadIdx.x] = acc;
    __syncthreads();
    for (int s = 128; s > 0; s >>= 1) {
        if ((int)threadIdx.x < s) red[threadIdx.x] += red[threadIdx.x + s];
        __syncthreads();
    }
    if (threadIdx.x == 0) partial_sum[blockIdx.x] = red[0];

    __syncthreads();
    red[threadIdx.x] = cnt;
    __syncthreads();
    for (int s = 128; s > 0; s >>= 1) {
        if ((int)threadIdx.x < s) red[threadIdx.x] += red[threadIdx.x + s];
        __syncthreads();
    }
    if (threadIdx.x == 0) partial_cnt[blockIdx.x] = red[0];
}

// ---------------------------------------------------------------------------
// Stage 3: Huber + combine 256 partials -> (total, L_c, L_s). Single block.
// ---------------------------------------------------------------------------
__global__ __launch_bounds__(256) void scmpa_finalize(const float* __restrict__ pred,
                                                      const float* __restrict__ tgt,
                                                      const float* __restrict__ partial_sum,
                                                      const float* __restrict__ partial_cnt,
                                                      float* __restrict__ out) {
    __shared__ float red[256];
    __shared__ float tot_s, cnt_s;

    // pair-loss total
    red[threadIdx.x] = partial_sum[threadIdx.x];
    __syncthreads();
    for (int s = 128; s > 0; s >>= 1) {
        if ((int)threadIdx.x < s) red[threadIdx.x] += red[threadIdx.x + s];
        __syncthreads();
    }
    if (threadIdx.x == 0) tot_s = red[0];
    __syncthreads();

    // pair count
    red[threadIdx.x] = partial_cnt[threadIdx.x];
    __syncthreads();
    for (int s = 128; s > 0; s >>= 1) {
        if ((int)threadIdx.x < s) red[threadIdx.x] += red[threadIdx.x + s];
        __syncthreads();
    }
    if (threadIdx.x == 0) cnt_s = red[0];
    __syncthreads();

    // Huber
    float h = 0.0f;
    for (int t = threadIdx.x; t < NROWS * 4; t += 256) {
        const int r = t >> 2;
        const int c = t & 3;
        const float d  = pred[t] - tgt[(size_t)r * 5 + c];
        const float ad = fabsf(d);
        h += (ad <= HUBER_DELTA_F) ? 0.5f * d * d
                                   : HUBER_DELTA_F * (ad - 0.5f * HUBER_DELTA_F);
    }
    red[threadIdx.x] = h;
    __syncthreads();
    for (int s = 128; s > 0; s >>= 1) {
        if ((int)threadIdx.x < s) red[threadIdx.x] += red[threadIdx.x + s];
        __syncthreads();
    }

    if (threadIdx.x == 0) {
        const float L_s = red[0] / (float)(NROWS * 4);
        const float L_c = (cnt_s > 0.0f) ? (tot_s / fmaxf(cnt_s, 1.0f)) : 0.0f;
        out[0] = L_c + L_s;   // LAMBDA_C = LAMBDA_S = 1
        out[1] = L_c;
        out[2] = L_s;
    }
}

extern "C" void kernel_launch(void* const* d_in, const int* in_sizes, int n_in,
                              void* d_out, int out_size, void* d_ws, size_t ws_size,
                              hipStream_t stream) {
    (void)in_sizes; (void)n_in; (void)out_size; (void)ws_size;
    const float* zref  = (const float*)d_in[0];
    const float* zperf = (const float*)d_in[1];
    const float* pred  = (const float*)d_in[2];
    const float* tgt   = (const float*)d_in[3];

    // workspace layout (floats): sim | pdot | pna | pnb | psum | pcnt
    float* ws   = (float*)d_ws;
    float* sim  = ws;                               // 4096
    float* pdot = sim + NROWS;                      // 8*4096
    float* pna  = pdot + (size_t)KSPLIT * NROWS;    // 8*4096
    float* pnb  = pna + (size_t)KSPLIT * NROWS;     // 8*4096
    float* psum = pnb + (size_t)KSPLIT * NROWS;     // 256
    float* pcnt = psum + 256;                       // 256

    const int blocks1 = (NROWS / (WAVES_PER_BLOCK * ROWS_PER_WAVE)) * KSPLIT;  // 256
    scmpa_partials_wmma<<<blocks1, 256, 0, stream>>>(zref, zperf, pdot, pna, pnb);
    scmpa_sim_combine<<<NROWS / 256, 256, 0, stream>>>(pdot, pna, pnb, sim);
    scmpa_pairwise<<<256, 256, 0, stream>>>(sim, tgt, psum, pcnt);
    scmpa_finalize<<<1, 256, 0, stream>>>(pred, tgt, psum, pcnt, (float*)d_out);
}